// Grid_40132174414020
// MI455X (gfx1250) — compile-verified
//
#include <hip/hip_runtime.h>

// ---------------------------------------------------------------------------
// MPM particle-to-grid scatter for MI455X (gfx1250, wave32).
//
// Bottleneck: 32M scattered no-return f32 atomic adds into a 32MB grid that
// lives entirely in the 192MB L2 -> L2 atomic throughput bound. Inputs are
// 28MB streamed once (~1.2us @ 23.3TB/s). No contraction exists anywhere in
// this workload, so the matrix (WMMA) pipe is intentionally unused; the win
// is on the memory side: native global_atomic_add_f32 (no CAS loop, no
// return -> STOREcnt only), th:NT loads for the once-streamed particle data
// so it doesn't evict the grid working set from L2, and global_prefetch_b8.
//
// p2g_scatter is defined FIRST so the disasm snippet shows the hot kernel.
// ---------------------------------------------------------------------------

#define GRID_N    128
#define NUM_CELLS (GRID_N * GRID_N * GRID_N)
#define INV_CELL  64.0f

// Fast no-return device-memory f32 atomic (native global_atomic_add_f32 on
// CDNA; d_out is coarse-grained device memory so the "unsafe" variant is the
// correct choice).
__device__ __forceinline__ void gatomic_add(float* p, float v) {
  unsafeAtomicAdd(p, v);
}

__global__ __launch_bounds__(256) void p2g_scatter(
    const float* __restrict__ pos,   // [N,3]
    const float* __restrict__ vel,   // [N,3]
    const float* __restrict__ mass,  // [N]
    float* __restrict__ out,         // [NUM_CELLS,4] = {mass, mom_xyz}
    int n) {
  const int p = blockIdx.x * blockDim.x + threadIdx.x;
  if (p >= n) return;

  // Run-ahead prefetch on the particle streams (global_prefetch_b8).
  const int pf = p + 8192;
  if (pf < n) {
    __builtin_prefetch(pos + 3 * pf, 0, 1);
    __builtin_prefetch(vel + 3 * pf, 0, 1);
  }

  // Streamed-once inputs: non-temporal so they don't displace the 32MB grid
  // accumulator working set in L2.
  const float px = __builtin_nontemporal_load(pos + 3 * p + 0);
  const float py = __builtin_nontemporal_load(pos + 3 * p + 1);
  const float pz = __builtin_nontemporal_load(pos + 3 * p + 2);
  const float vx = __builtin_nontemporal_load(vel + 3 * p + 0);
  const float vy = __builtin_nontemporal_load(vel + 3 * p + 1);
  const float vz = __builtin_nontemporal_load(vel + 3 * p + 2);
  const float m  = __builtin_nontemporal_load(mass + p);

  // Position in cell units; base cell + fractional part.
  const float rx = px * INV_CELL, ry = py * INV_CELL, rz = pz * INV_CELL;
  const float bx = floorf(rx),   by = floorf(ry),   bz = floorf(rz);
  const int   ix = (int)bx,      iy = (int)by,      iz = (int)bz;
  const float fx = rx - bx,      fy = ry - by,      fz = rz - bz;

  // Trilinear weights: corner offset 0 -> (1-f), offset 1 -> f.
  const float wx0 = 1.0f - fx, wx1 = fx;
  const float wy0 = 1.0f - fy, wy1 = fy;
  // Fold particle mass into the z-pair: each corner weight is then 1 mul.
  const float zm0 = (1.0f - fz) * m, zm1 = fz * m;

  const float xy00 = wx0 * wy0, xy01 = wx0 * wy1;
  const float xy10 = wx1 * wy0, xy11 = wx1 * wy1;

  // hash = z + x*128 + y*128*128
  const int h000 = iz + (ix << 7) + (iy << 14);

#define CORNER(DI, DJ, DK, XY, ZM)                                          \
  {                                                                         \
    const int h = h000 + (DI) * GRID_N + (DJ) * (GRID_N * GRID_N) + (DK);   \
    if (h >= 0 && h < NUM_CELLS) { /* matches reference validity mask */    \
      const float sm = (XY) * (ZM);                                         \
      float* c = out + (h << 2);                                            \
      gatomic_add(c + 0, sm);                                               \
      gatomic_add(c + 1, sm * vx);                                          \
      gatomic_add(c + 2, sm * vy);                                          \
      gatomic_add(c + 3, sm * vz);                                          \
    }                                                                       \
  }

  CORNER(0, 0, 0, xy00, zm0)
  CORNER(0, 0, 1, xy00, zm1)
  CORNER(0, 1, 0, xy01, zm0)
  CORNER(0, 1, 1, xy01, zm1)
  CORNER(1, 0, 0, xy10, zm0)
  CORNER(1, 0, 1, xy10, zm1)
  CORNER(1, 1, 0, xy11, zm0)
  CORNER(1, 1, 1, xy11, zm1)
#undef CORNER
}

// Bulk zero via 128-bit stores; no in-kernel tail loop (tail handled by a
// separate tiny launch from the host wrapper).
__global__ __launch_bounds__(256) void zero4(float4* __restrict__ out4, int n4) {
  const int i = blockIdx.x * blockDim.x + threadIdx.x;
  if (i < n4) out4[i] = make_float4(0.f, 0.f, 0.f, 0.f);
}

__global__ void zero1(float* __restrict__ out, int n) {
  const int i = (int)threadIdx.x;
  if (i < n) out[i] = 0.f;
}

extern "C" void kernel_launch(void* const* d_in, const int* in_sizes, int n_in,
                              void* d_out, int out_size, void* d_ws, size_t ws_size,
                              hipStream_t stream) {
  (void)n_in; (void)d_ws; (void)ws_size;
  const float* pos  = (const float*)d_in[0];  // [N,3]
  const float* vel  = (const float*)d_in[1];  // [N,3]
  const float* mass = (const float*)d_in[2];  // [N]
  float* out = (float*)d_out;                 // [NUM_CELLS,4]
  const int n = in_sizes[2];

  // Accumulator must start at zero every call (harness poisons once; graph
  // replays do not re-zero anything).
  const int n4  = out_size >> 2;
  const int rem = out_size & 3;
  if (n4 > 0) {
    const int blocks = (n4 + 255) / 256;
    zero4<<<blocks, 256, 0, stream>>>((float4*)out, n4);
  }
  if (rem > 0) {
    zero1<<<1, rem, 0, stream>>>(out + (out_size & ~3), rem);
  }

  const int blocks = (n + 255) / 256;
  p2g_scatter<<<blocks, 256, 0, stream>>>(pos, vel, mass, out, n);
}